// compress_jpeg_40114994545190
// MI455X (gfx1250) — compile-verified
//
#include <hip/hip_runtime.h>

typedef __attribute__((ext_vector_type(2))) float v2f;
typedef __attribute__((ext_vector_type(8))) float v8f;

#define IMG_H 512
#define IMG_W 512
#define NBATCH 32

// Padded LDS strides (even -> 8B-aligned v2f pairs at [mn*stride + even_k])
#define YSTR 34   // 32x32 Y tile, column-major
#define CSTR 18   // 16x16 chroma / basis / scratch tiles

// DCT + quantize one 16x16 tile (four 8x8 blocks) with one wave32.
// s:   LDS tile, COLUMN-major (s[n*stride + k] = S(k,n)), values before -128
// m1:  per-wave 16x18 LDS scratch (row-major, padded)
// G:   16x18 LDS: G[i*CSTR + j] = Td(j,i)  (transposed block-diag basis, zeros explicit)
__device__ __forceinline__ void dct_tile(const float* __restrict__ s, int s_stride,
                                         float* __restrict__ m1,
                                         const float* __restrict__ G,
                                         const float* __restrict__ scale,
                                         const float* __restrict__ qtab,
                                         float* __restrict__ outp,
                                         int nbw, int row0, int col0) {
    const int lane = threadIdx.x & 31;
    const int half = lane >> 4;    // K half for A/B fragments, M half for C/D
    const int mn   = lane & 15;    // M (A) / N (B,C,D) index

    // ---- pass 1: acc = Td^T * (S - 128)  (four K=4 WMMA steps) ----
    v8f acc = {};
#pragma unroll
    for (int kk = 0; kk < 4; ++kk) {
        const int kb = kk * 4 + half * 2;                       // even
        const v2f a = *(const v2f*)(G + mn * CSTR + kb);        // Td(kb..kb+1, mn)
        v2f b = *(const v2f*)(s + mn * s_stride + kb);          // S(kb..kb+1, mn)
        b.x -= 128.0f;
        b.y -= 128.0f;
        acc = __builtin_amdgcn_wmma_f32_16x16x4_f32(false, a, false, b,
                                                    (short)0, acc, false, false);
    }

    // ---- transpose pass-1 result through per-wave LDS scratch ----
#pragma unroll
    for (int r = 0; r < 8; ++r)
        m1[(r + half * 8) * CSTR + mn] = acc[r];   // C/D layout -> row-major

    __builtin_amdgcn_wave_barrier();
    asm volatile("s_wait_dscnt 0" ::: "memory"); // same-wave LDS RAW
    __builtin_amdgcn_wave_barrier();

    // ---- pass 2: acc2 = M1 * Td ----
    v8f acc2 = {};
#pragma unroll
    for (int kk = 0; kk < 4; ++kk) {
        const int kb = kk * 4 + half * 2;
        const v2f a = *(const v2f*)(m1 + mn * CSTR + kb);       // M1(mn, kb..kb+1)
        const v2f b = *(const v2f*)(G + mn * CSTR + kb);        // Td(kb..kb+1, mn)
        acc2 = __builtin_amdgcn_wmma_f32_16x16x4_f32(false, a, false, b,
                                                     (short)0, acc2, false, false);
    }

    // ---- scale / quantize / round-half-even / store in block-split layout ----
#pragma unroll
    for (int r = 0; r < 8; ++r) {
        const int tu = r + half * 8;
        const int tv = mn;
        const int prow = row0 + tu, pcol = col0 + tv;
        const int u = prow & 7, v = pcol & 7;
        const int blk = (prow >> 3) * nbw + (pcol >> 3);
        const float val = acc2[r] * scale[u * 8 + v] / qtab[u * 8 + v];
        outp[(size_t)blk * 64 + u * 8 + v] = rintf(val);
    }
}

__global__ __launch_bounds__(256) void compress_jpeg_kernel(
    const float* __restrict__ image, const float* __restrict__ matrix,
    const float* __restrict__ shift, const float* __restrict__ dct_tensor,
    const float* __restrict__ dct_scale, const float* __restrict__ y_table,
    const float* __restrict__ c_table, float* __restrict__ out) {

    __shared__ float yt[32 * YSTR];      // Y tile, column-major, padded
    __shared__ float cbt[16 * CSTR];     // pooled Cb tile, column-major
    __shared__ float crt[16 * CSTR];     // pooled Cr tile, column-major
    __shared__ float G[16 * CSTR];       // G[i][j] = Td(j,i), zeros explicit
    __shared__ float m1[6][16 * CSTR];   // per-wave transpose scratch

    const int tid   = threadIdx.x;
    const int tileX = blockIdx.x, tileY = blockIdx.y, b = blockIdx.z;

    // Build transposed block-diagonal basis with explicit zeros (1 elem/thread).
    // t[x][u] = dct_tensor[x][0][u][0]  (t[0][0] == 1)
    {
        const int i = tid >> 4, j = tid & 15;   // G[i][j] = Td(j,i)
        G[i * CSTR + j] = ((i ^ j) & 8) ? 0.0f
                        : dct_tensor[(j & 7) * 512 + (i & 7) * 8];
    }

    // uniform 3x3 color matrix + shift (scalar loads)
    const float m00 = matrix[0], m01 = matrix[1], m02 = matrix[2];
    const float m10 = matrix[3], m11 = matrix[4], m12 = matrix[5];
    const float m20 = matrix[6], m21 = matrix[7], m22 = matrix[8];
    const float s0 = shift[0], s1 = shift[1], s2 = shift[2];

    // ---- stage 1: one 2x2 quad per thread: RGB->YCbCr + chroma pooling ----
    const int qr = tid >> 4, qc = tid & 15;
    const int py = tileY * 32 + qr * 2;
    const int px = tileX * 32 + qc * 2;          // even -> float2-aligned
    const float* imgb = image + (size_t)b * 3 * IMG_H * IMG_W;

    float cbs = 0.f, crs = 0.f;
#pragma unroll
    for (int dy = 0; dy < 2; ++dy) {
        const size_t off = (size_t)(py + dy) * IMG_W + px;
        const v2f rp = *(const v2f*)(imgb + off);
        const v2f gp = *(const v2f*)(imgb + (size_t)IMG_H * IMG_W + off);
        const v2f bp = *(const v2f*)(imgb + (size_t)2 * IMG_H * IMG_W + off);
#pragma unroll
        for (int dx = 0; dx < 2; ++dx) {
            const float r  = rp[dx] * 255.0f;
            const float g  = gp[dx] * 255.0f;
            const float bl = bp[dx] * 255.0f;
            const float Y  = r * m00 + g * m10 + bl * m20 + s0;
            const float Cb = r * m01 + g * m11 + bl * m21 + s1;
            const float Cr = r * m02 + g * m12 + bl * m22 + s2;
            yt[(qc * 2 + dx) * YSTR + (qr * 2 + dy)] = Y;   // column-major
            cbs += Cb;
            crs += Cr;
        }
    }
    cbt[qc * CSTR + qr] = cbs * 0.25f;   // column-major (col=qc, row=qr)
    crt[qc * CSTR + qr] = crs * 0.25f;

    __syncthreads();

    // ---- stage 2: per-wave WMMA DCT + quantization ----
    const int wave = tid >> 5;
    float* y_out  = out;
    float* cb_out = out + (size_t)NBATCH * 4096 * 64;
    float* cr_out = cb_out + (size_t)NBATCH * 1024 * 64;

    if (wave < 4) {
        const int sy = (wave >> 1) * 16, sx = (wave & 1) * 16;  // both even
        dct_tile(&yt[sx * YSTR + sy], YSTR, m1[wave], G, dct_scale, y_table,
                 y_out + (size_t)b * 4096 * 64, IMG_W / 8,
                 tileY * 32 + sy, tileX * 32 + sx);
    } else if (wave == 4) {
        dct_tile(cbt, CSTR, m1[4], G, dct_scale, c_table,
                 cb_out + (size_t)b * 1024 * 64, (IMG_W / 2) / 8,
                 tileY * 16, tileX * 16);
    } else if (wave == 5) {
        dct_tile(crt, CSTR, m1[5], G, dct_scale, c_table,
                 cr_out + (size_t)b * 1024 * 64, (IMG_W / 2) / 8,
                 tileY * 16, tileX * 16);
    }
    // waves 6,7 idle in stage 2 (no further block-wide barriers)
}

extern "C" void kernel_launch(void* const* d_in, const int* in_sizes, int n_in,
                              void* d_out, int out_size, void* d_ws, size_t ws_size,
                              hipStream_t stream) {
    (void)in_sizes; (void)n_in; (void)out_size; (void)d_ws; (void)ws_size;
    dim3 grid(IMG_W / 32, IMG_H / 32, NBATCH);   // 16 x 16 x 32
    dim3 block(256);                             // 8 wave32s
    compress_jpeg_kernel<<<grid, block, 0, stream>>>(
        (const float*)d_in[0], (const float*)d_in[1], (const float*)d_in[2],
        (const float*)d_in[3], (const float*)d_in[4], (const float*)d_in[5],
        (const float*)d_in[6], (float*)d_out);
}